// SRenderY_53480932770033
// MI455X (gfx1250) — compile-verified
//
#include <hip/hip_runtime.h>
#include <stdint.h>

// Problem constants from the reference
#define N_ 16
#define V_ 5023
#define F_ 9976
#define H_ 512
#define W_ 512

constexpr int NPIX   = N_ * H_ * W_;   // 4,194,304 pixels (K=1)
constexpr int NQUAD  = NPIX / 4;       // 1,048,576
constexpr int NFACE3 = F_ * 3;         // 29,928 dwords = 119,712 B (fits in 320 KB LDS)
constexpr int NZ     = N_ * V_;        // 80,368 z-column elements

typedef int v4i_cdna5 __attribute__((ext_vector_type(4)));

// ---------------------------------------------------------------------------
// CDNA5 async global->LDS copy (ASYNCcnt path). Builtin confirmed present:
// signature wants (v4i AS1*, v4i AS3*, imm offset, imm cpol).
// ---------------------------------------------------------------------------
__device__ __forceinline__ void async_g2l_b128(int* g, int* l) {
#if __has_builtin(__builtin_amdgcn_global_load_async_to_lds_b128)
  v4i_cdna5* gv = (v4i_cdna5*)g;  // generic ptr, pointee type change only
  v4i_cdna5* lv = (v4i_cdna5*)l;
  __builtin_amdgcn_global_load_async_to_lds_b128(
      (__attribute__((address_space(1))) v4i_cdna5*)gv,
      (__attribute__((address_space(3))) v4i_cdna5*)lv,
      0, 0);
#else
  unsigned int       lofs = (unsigned int)(unsigned long long)l; // LDS byte offset
  unsigned long long ga   = (unsigned long long)g;               // 64-bit global address
  asm volatile("global_load_async_to_lds_b128 %0, %1, off"
               :: "v"(lofs), "v"(ga)
               : "memory");
#endif
}

__device__ __forceinline__ void wait_async0() {
#if __has_builtin(__builtin_amdgcn_s_wait_asynccnt)
  __builtin_amdgcn_s_wait_asynccnt(0);
#else
  asm volatile("s_wait_asynccnt 0x0" ::: "memory");
#endif
}

// ---------------------------------------------------------------------------
// Kernel A: global min/max over tv[:,:,2]  ->  ws[0]=tmax, ws[1]=1/(tmax-tmin)
// ---------------------------------------------------------------------------
__global__ __launch_bounds__(1024) void srender_minmax_kernel(
    const float* __restrict__ tv, float* __restrict__ ws) {
  __shared__ float smin[1024];
  __shared__ float smax[1024];
  const int t = threadIdx.x;
  float mn =  3.402823466e38f;
  float mx = -3.402823466e38f;
  for (int i = t; i < NZ; i += 1024) {
    const float v = tv[i * 3 + 2];
    mn = fminf(mn, v);
    mx = fmaxf(mx, v);
  }
  smin[t] = mn;
  smax[t] = mx;
  __syncthreads();
  for (int s = 512; s > 0; s >>= 1) {
    if (t < s) {
      smin[t] = fminf(smin[t], smin[t + s]);
      smax[t] = fmaxf(smax[t], smax[t + s]);
    }
    __syncthreads();
  }
  if (t == 0) {
    const float tmin = smin[0];
    const float tmax = smax[0];
    ws[0] = tmax;
    ws[1] = 1.0f / (tmax - tmin);
  }
}

// ---------------------------------------------------------------------------
// Kernel B: per-pixel depth = mask ? 0 : sum_c bary_c * (tmax - t[nb, faces[f,c]]) * inv
// ---------------------------------------------------------------------------
__device__ __forceinline__ float pixel_val(int idx, float b0, float b1, float b2,
                                           float tmax, float inv,
                                           const float* __restrict__ tv,
                                           const int* sfaces) {
  const bool valid = idx >= 0;
  const int  ic    = valid ? idx : 0;        // reference clamps masked idx to 0
  const int  f     = ic % F_;
  const int  nb    = ic / F_;
  const int  base  = nb * V_;
  const int  v0    = sfaces[3 * f + 0];
  const int  v1    = sfaces[3 * f + 1];
  const int  v2    = sfaces[3 * f + 2];
  const float t0 = tv[(base + v0) * 3 + 2];
  const float t1 = tv[(base + v1) * 3 + 2];
  const float t2 = tv[(base + v2) * 3 + 2];
  const float r  = (b0 * (tmax - t0) + b1 * (tmax - t1) + b2 * (tmax - t2)) * inv;
  return valid ? r : 0.0f;
}

__global__ __launch_bounds__(256) void srender_depth_kernel(
    const float* __restrict__ tv, const float* __restrict__ bary,
    const int* __restrict__ faces, const int* __restrict__ p2f,
    const float* __restrict__ wsf, float* __restrict__ out) {
  __shared__ int sfaces[NFACE3];  // 119,712 B, staged once per block via async DMA

  // Stage the faces index table global->LDS with async b128 copies.
  int* gfaces = const_cast<int*>(faces);
  for (int i = threadIdx.x; i < NFACE3 / 4; i += 256) {
    async_g2l_b128(gfaces + 4 * i, sfaces + 4 * i);
  }
  wait_async0();
  __syncthreads();

  const float tmax = wsf[0];
  const float inv  = wsf[1];

  const int stride = gridDim.x * 256;
  for (int q = blockIdx.x * 256 + threadIdx.x; q < NQUAD; q += stride) {
    const int4   i4 = ((const int4*)p2f)[q];
    const float4 A  = ((const float4*)bary)[3 * q + 0];
    const float4 B  = ((const float4*)bary)[3 * q + 1];
    const float4 C  = ((const float4*)bary)[3 * q + 2];
    float4 r;
    r.x = pixel_val(i4.x, A.x, A.y, A.z, tmax, inv, tv, sfaces);
    r.y = pixel_val(i4.y, A.w, B.x, B.y, tmax, inv, tv, sfaces);
    r.z = pixel_val(i4.z, B.z, B.w, C.x, tmax, inv, tv, sfaces);
    r.w = pixel_val(i4.w, C.y, C.z, C.w, tmax, inv, tv, sfaces);
    ((float4*)out)[q] = r;
  }
}

// ---------------------------------------------------------------------------
// Launch
// ---------------------------------------------------------------------------
extern "C" void kernel_launch(void* const* d_in, const int* in_sizes, int n_in,
                              void* d_out, int out_size, void* d_ws, size_t ws_size,
                              hipStream_t stream) {
  const float* tv    = (const float*)d_in[0];  // (N,V,3) f32
  const float* bary  = (const float*)d_in[1];  // (N,H,W,1,3) f32
  const int*   faces = (const int*)d_in[2];    // (F,3) i32
  const int*   p2f   = (const int*)d_in[3];    // (N,H,W,1) i32
  float*       wsf   = (float*)d_ws;           // 8 bytes used: tmax, inv_range
  float*       out   = (float*)d_out;          // (N,1,H,W) f32

  srender_minmax_kernel<<<dim3(1), dim3(1024), 0, stream>>>(tv, wsf);
  srender_depth_kernel<<<dim3(512), dim3(256), 0, stream>>>(tv, bary, faces, p2f, wsf, out);
}